// BackWarp_2130303779365
// MI455X (gfx1250) — compile-verified
//
#include <hip/hip_runtime.h>

// dense_image_warp (tfa-style bilinear backward warp), B=32, H=512, W=512, C=4.
// Memory-bound: ~320 MB min traffic -> ~14us at 23.3 TB/s; FLOPs are negligible.
// Strategy: one thread per output pixel (grid-stride), float4 (B128) vectorized
// gathers of the 4 corners, NT-hinted streaming of flow/output so the 192MB L2
// is reserved for the reused image data, and gfx1250 global_prefetch on the
// next iteration's flow.

typedef float v2f __attribute__((ext_vector_type(2)));
typedef float v4f __attribute__((ext_vector_type(4)));

namespace {
constexpr int kB = 32;
constexpr int kH = 512;
constexpr int kW = 512;
constexpr int kNPix = kB * kH * kW;          // 8,388,608 pixels
constexpr int kThreads = 256;                // 8 wave32s per block
constexpr int kBlocks = 8192;                // ~4 grid-stride iterations/thread
}  // namespace

__global__ __launch_bounds__(kThreads) void warp_bilinear_kernel(
    const v4f* __restrict__ image,   // [B*H*W] float4 (C=4 -> 16B aligned)
    const v2f* __restrict__ flow,    // [B*H*W] float2 (flow_y, flow_x)
    v4f* __restrict__ out) {         // [B*H*W] float4
  int idx = blockIdx.x * kThreads + threadIdx.x;
  const int stride = gridDim.x * kThreads;

  for (; idx < kNPix; idx += stride) {
    // gfx1250 global_prefetch_b8 of the next iteration's flow (speculative;
    // OOB addresses are silently dropped per CDNA5 ISA).
    __builtin_prefetch(reinterpret_cast<const char*>(flow + idx) +
                           (size_t)stride * sizeof(v2f),
                       /*rw=*/0, /*locality=*/1);

    // Flow is streamed exactly once -> non-temporal load (keep L2 for image).
    v2f f = __builtin_nontemporal_load(flow + idx);

    // W = 512, H = 512 -> pure shift/mask index decomposition.
    const int j = idx & (kW - 1);          // x
    const int i = (idx >> 9) & (kH - 1);   // y
    const int b = idx >> 18;               // batch

    // Query point: (i - flow_y, j - flow_x).
    const float qy = (float)i - f.x;
    const float qx = (float)j - f.y;

    // Edge-clamped floors and alphas (matches tfa interpolate_bilinear).
    float fy = fminf(fmaxf(floorf(qy), 0.0f), (float)(kH - 2));
    float fx = fminf(fmaxf(floorf(qx), 0.0f), (float)(kW - 2));
    const float ay = fminf(fmaxf(qy - fy, 0.0f), 1.0f);
    const float ax = fminf(fmaxf(qx - fx, 0.0f), 1.0f);
    const int iy = (int)fy;
    const int ix = (int)fx;

    // Four corner gathers: each is a single 16B global_load_b128.
    const v4f* img = image + ((size_t)b * (kH * kW)) + ((size_t)iy * kW) + ix;
    const v4f tl = img[0];
    const v4f tr = img[1];
    const v4f bl = img[kW];
    const v4f br = img[kW + 1];

    // Bilinear blend (vector FMAs over the 4 channels).
    const v4f top = tl + ax * (tr - tl);
    const v4f bot = bl + ax * (br - bl);
    const v4f res = top + ay * (bot - top);

    // Output is streamed once -> non-temporal b128 store.
    __builtin_nontemporal_store(res, out + idx);
  }
}

extern "C" void kernel_launch(void* const* d_in, const int* in_sizes, int n_in,
                              void* d_out, int out_size, void* d_ws,
                              size_t ws_size, hipStream_t stream) {
  (void)in_sizes; (void)n_in; (void)d_ws; (void)ws_size; (void)out_size;
  const v4f* image = reinterpret_cast<const v4f*>(d_in[0]);  // [32,512,512,4] f32
  const v2f* flow  = reinterpret_cast<const v2f*>(d_in[1]);  // [32,512,512,2] f32
  v4f* out = reinterpret_cast<v4f*>(d_out);                  // [32,512,512,4] f32

  warp_bilinear_kernel<<<kBlocks, kThreads, 0, stream>>>(image, flow, out);
}